// GrowingShapeUpToMaxPedestrians_37271726195503
// MI455X (gfx1250) — compile-verified
//
#include <hip/hip_runtime.h>
#include <math.h>

typedef __attribute__((ext_vector_type(2))) float v2f;
typedef __attribute__((ext_vector_type(8))) float v8f;

#define N_BINS     512
#define BIN_SCALE2 2.0f      // bins over d^2 in [0,256), width 0.5
#define MAX_PED    32
#define ROW_CAP    192
#define ROWS_WG    16
#define THREADS    256
#define WAVES      8

// Per-pedestrian precompute: |p|^2 and heading half-plane coefficients.
__global__ void prep_kernel(const float2* __restrict__ past,
                            const float2* __restrict__ pos,
                            float* __restrict__ n2,
                            float* __restrict__ hx,
                            float* __restrict__ hy,
                            float* __restrict__ hc,
                            int n)
{
    int i = blockIdx.x * blockDim.x + threadIdx.x;
    if (i >= n) return;
    float x = pos[i].x, y = pos[i].y;
    n2[i] = x * x + y * y;
    float dx = x - past[i].x;
    float dy = y - past[i].y;
    if (dx == 0.0f && dy == 0.0f) { dx = 1.0f; dy = 0.0f; }  // atan2(0,0)=0 -> heading +x
    hx[i] = dx; hy[i] = dy;
    hc[i] = -(dx * x + dy * y);
}

// Monotone partition of d^2; only pass1/pass2 consistency matters, no sqrt needed.
__device__ __forceinline__ int bin_of_d2(float d2)
{
    float t = d2 * BIN_SCALE2;
    t = t > 511.0f ? 511.0f : t;
    int b = (int)t;
    return b < 0 ? 0 : b;
}

// k-th smallest (0-indexed) among d^2 values of v[] that fall in `bin`, tie-aware.
__device__ float kth_in_bin(const float* v, int m, int bin, int k)
{
    for (int a = 0; a < m; ++a) {
        float va = v[a];
        if (bin_of_d2(va) != bin) continue;
        int lo = 0, eq = 0;
        for (int b = 0; b < m; ++b) {
            float vb = v[b];
            if (bin_of_d2(vb) != bin) continue;
            lo += (vb <  va);
            eq += (vb == va);
        }
        if (lo <= k && k < lo + eq) return va;
    }
    return 256.0f;  // overflow fallback (capacity exceeded; statistically unreachable)
}

__launch_bounds__(THREADS)
__global__ void radii_kernel(const float2* __restrict__ pos,
                             const float* __restrict__ n2,
                             const float* __restrict__ hx,
                             const float* __restrict__ hy,
                             const float* __restrict__ hc,
                             float* __restrict__ out,
                             int n)
{
    __shared__ unsigned int hist[ROWS_WG][N_BINS];
    __shared__ float        buf [ROWS_WG][ROW_CAP];
    __shared__ unsigned int cnt [ROWS_WG];
    __shared__ int          sB31[ROWS_WG], sB32[ROWS_WG];
    __shared__ int          sR31[ROWS_WG], sR32[ROWS_WG];
    __shared__ int          sDone[ROWS_WG];

    const int  tid  = threadIdx.x;
    const int  lane = tid & 31;
    const int  wave = tid >> 5;
    const int  col  = lane & 15;
    const bool hi   = lane >= 16;
    const int  i0   = blockIdx.x * ROWS_WG;
    const int  nTiles = n >> 4;

    for (int b = tid; b < ROWS_WG * N_BINS; b += THREADS)
        (&hist[0][0])[b] = 0u;
    __syncthreads();

    // A operands (16x4 f32: lanes 0-15 = K0/K1, lanes 16-31 = K2/K3), fixed per row-block.
    const int   irow = i0 + col;
    const float px = pos[irow].x, py = pos[irow].y;
    const float pn2 = n2[irow];
    const float fx = hx[irow], fy = hy[irow], fc = hc[irow];
    v2f aD, aF;
    aD[0] = hi ? pn2 : px;   aD[1] = hi ? 1.0f : py;     // dist^2 GEMM: [x,y,|p|^2,1]
    aF[0] = hi ? fc  : fx;   aF[1] = hi ? 0.0f : fy;     // FOV GEMM:   [hx,hy,-h.p,0]

    const v8f czero = {};

    // ---- pass 1: per-row 512-bin histogram over in-sight squared distances ----
    for (int t = wave; t < nTiles; t += WAVES) {
        const int   j   = (t << 4) + col;
        const float xj  = pos[j].x, yj = pos[j].y;
        const float jn2 = n2[j];
        v2f bD, bF;  // B 4x16 f32: lanes 0-15 = K0/K1 rows, lanes 16-31 = K2/K3 rows
        bD[0] = hi ? 1.0f : -2.0f * xj;  bD[1] = hi ? jn2  : -2.0f * yj;
        bF[0] = hi ? 1.0f : xj;          bF[1] = hi ? 0.0f : yj;

        v8f d2 = __builtin_amdgcn_wmma_f32_16x16x4_f32(false, aD, false, bD,
                                                       (short)0, czero, false, false);
        v8f fv = __builtin_amdgcn_wmma_f32_16x16x4_f32(false, aF, false, bF,
                                                       (short)0, czero, false, false);
        #pragma unroll
        for (int v = 0; v < 8; ++v) {
            const int M  = hi ? (v + 8) : v;
            const int gi = i0 + M;
            if ((fv[v] > 0.0f) && (gi != j))
                atomicAdd(&hist[M][bin_of_d2(d2[v])], 1u);
        }
    }
    __syncthreads();

    // ---- scan: locate bins holding order statistics 31 and 32 ----
    if (tid < ROWS_WG) {
        unsigned int cum = 0;
        int b31 = N_BINS - 1, b32 = N_BINS - 1, r31 = 0, r32 = 0;
        bool got31 = false, got32 = false;
        for (int b = 0; b < N_BINS; ++b) {
            unsigned int h  = hist[tid][b];
            unsigned int nc = cum + h;
            if (!got31 && nc >= 32u) { b31 = b; r31 = 31 - (int)cum; got31 = true; }
            if (!got32 && nc >= 33u) { b32 = b; r32 = 32 - (int)cum; got32 = true; }
            cum = nc;
        }
        int done = 0;
        if (cum <= (unsigned)MAX_PED)  { out[i0 + tid] = 8.0f; done = 1; }
        else if (b31 >= 128)           { out[i0 + tid] = 8.0f; done = 1; }  // d31>=8 -> clip
        sB31[tid] = b31; sB32[tid] = b32;
        sR31[tid] = r31; sR32[tid] = r32;
        sDone[tid] = done;
        cnt[tid] = 0u;
    }
    __syncthreads();

    unsigned int needMask = 0;
    int tb31[8], tb32[8];
    #pragma unroll
    for (int v = 0; v < 8; ++v) {
        const int M = hi ? (v + 8) : v;
        tb31[v] = sB31[M]; tb32[v] = sB32[M];
    }
    for (int r = 0; r < ROWS_WG; ++r)
        needMask |= (sDone[r] ? 0u : 1u) << r;

    // ---- pass 2: gather exact d^2 values landing in the two target bins ----
    if (needMask != 0u) {  // uniform branch: EXEC stays all-ones for WMMA
        for (int t = wave; t < nTiles; t += WAVES) {
            const int   j   = (t << 4) + col;
            const float xj  = pos[j].x, yj = pos[j].y;
            const float jn2 = n2[j];
            v2f bD, bF;
            bD[0] = hi ? 1.0f : -2.0f * xj;  bD[1] = hi ? jn2  : -2.0f * yj;
            bF[0] = hi ? 1.0f : xj;          bF[1] = hi ? 0.0f : yj;

            v8f d2 = __builtin_amdgcn_wmma_f32_16x16x4_f32(false, aD, false, bD,
                                                           (short)0, czero, false, false);
            v8f fv = __builtin_amdgcn_wmma_f32_16x16x4_f32(false, aF, false, bF,
                                                           (short)0, czero, false, false);
            #pragma unroll
            for (int v = 0; v < 8; ++v) {
                const int M  = hi ? (v + 8) : v;
                const int gi = i0 + M;
                if (!((needMask >> M) & 1u)) continue;
                if ((fv[v] > 0.0f) && (gi != j)) {
                    const int bb = bin_of_d2(d2[v]);
                    if (bb == tb31[v] || bb == tb32[v]) {
                        unsigned int k = atomicAdd(&cnt[M], 1u);
                        if (k < (unsigned)ROW_CAP) buf[M][k] = d2[v];
                    }
                }
            }
        }
    }
    __syncthreads();

    // ---- final selection: sqrt only on the two chosen values per row ----
    if (tid < ROWS_WG && !sDone[tid]) {
        int m = (int)cnt[tid]; if (m > ROW_CAP) m = ROW_CAP;
        float q31 = kth_in_bin(&buf[tid][0], m, sB31[tid], sR31[tid]);
        float q32 = kth_in_bin(&buf[tid][0], m, sB32[tid], sR32[tid]);
        float d31 = sqrtf(q31 < 0.0f ? 0.0f : q31); d31 = d31 > 16.0f ? 16.0f : d31;
        float d32 = sqrtf(q32 < 0.0f ? 0.0f : q32); d32 = d32 > 16.0f ? 16.0f : d32;
        float r = 0.5f * (d31 + d32);
        r = r < 0.5f ? 0.5f : (r > 8.0f ? 8.0f : r);
        out[i0 + tid] = r;
    }
}

extern "C" void kernel_launch(void* const* d_in, const int* in_sizes, int n_in,
                              void* d_out, int out_size, void* d_ws, size_t ws_size,
                              hipStream_t stream)
{
    const float2* past = (const float2*)d_in[0];  // past_ped_positions (N,2) f32
    const float2* pos  = (const float2*)d_in[1];  // ped_positions      (N,2) f32
    // d_in[2] (indexes) does not affect the forward output.
    const int n = in_sizes[1] / 2;

    float* n2 = (float*)d_ws;       // 4*N floats of scratch
    float* hx = n2 + n;
    float* hy = hx + n;
    float* hc = hy + n;

    prep_kernel<<<(n + 255) / 256, 256, 0, stream>>>(past, pos, n2, hx, hy, hc, n);
    radii_kernel<<<n / ROWS_WG, THREADS, 0, stream>>>(pos, n2, hx, hy, hc,
                                                      (float*)d_out, n);
}